// SoftDTW_41927470744116
// MI455X (gfx1250) — compile-verified
//
#include <hip/hip_runtime.h>
#include <stdint.h>

// Problem constants (match reference: B, N, M = 32, 512, 512; GAMMA=1; BANDWIDTH=0)
#define Bb   32
#define Nn   512
#define Mm   512
#define DMAX (Nn + Mm)       // last diagonal index = 1024
#define BIGF 1e30f

#ifndef __has_builtin
#define __has_builtin(x) 0
#endif

#if defined(__HIP_DEVICE_COMPILE__) && \
    __has_builtin(__builtin_amdgcn_tensor_load_to_lds) && \
    __has_builtin(__builtin_amdgcn_s_wait_tensorcnt)
#define UDTW_HAVE_TDM 1
#else
#define UDTW_HAVE_TDM 0
#endif

typedef unsigned int v4u __attribute__((ext_vector_type(4)));
typedef int          v8i __attribute__((ext_vector_type(8)));
typedef int          v4i __attribute__((ext_vector_type(4)));

#if UDTW_HAVE_TDM
// Low 32 bits of a flat shared-memory address are the LDS byte offset
// (aperture lives entirely in the high 32 bits per CDNA5 aperture check).
__device__ __forceinline__ uint32_t lds_off(const void* p) {
  return (uint32_t)(uintptr_t)p;
}

// Issue one TDM descriptor: gather L float32 elements with a stride of
// (Mm-1) elements -- i.e. one anti-diagonal of a row-major [Nn x Mm]
// matrix -- into contiguous LDS starting at lds_byte.
//   addr(y) = global_addr + 4*(0 + y*tensor_dim0_stride), y = 0..L-1
__device__ __forceinline__ void tdm_diag_load(const float* gptr,
                                              uint32_t lds_byte, int L) {
  uint64_t ga = (uint64_t)(uintptr_t)gptr;
  v4u g0;
  g0.x = 0x1u;                                    // count=1 (valid user D#)
  g0.y = lds_byte;                                // lds_addr [63:32]
  g0.z = (uint32_t)ga;                            // global_addr[31:0]
  g0.w = ((uint32_t)(ga >> 32) & 0x01FFFFFFu)     // global_addr[56:32]
         | 0x80000000u;                           // type=2 ("image") [127:126]
  const uint32_t L32 = (uint32_t)L;
  v8i g1;
  g1[0] = (int)(2u << 16);                        // wg_mask=0 | data_size=2 (4B)
  g1[1] = (int)(1u << 16);                        // tensor_dim0 = 1 (bits 63:48)
  g1[2] = (int)((L32 & 0xFFFFu) << 16);           // dim0 hi=0 | tensor_dim1 lo16
  g1[3] = (int)(((L32 >> 16) & 0xFFFFu)           // tensor_dim1 hi16
                | (1u << 16));                    // tile_dim0 = 1 (bits 127:112)
  g1[4] = (int)(L32 & 0xFFFFu);                   // tile_dim1 = L | tile_dim2 = 0
  g1[5] = (int)(Mm - 1);                          // tensor_dim0_stride[31:0] = 511
  g1[6] = 0;                                      // stride0 hi | stride1 lo = 0
  g1[7] = 0;                                      // stride1 hi = 0
  v4i g2 = {1, 1, 0, 0};                          // tensor_dim2=1, dim3=1 (unused)
  v4i g3 = {0, (int)(1u << 16), 0, 0};            // tensor_dim4=1 (unused)
  v8i g4 = {0, 0, 0, 0, 0, 0, 0, 0};              // extra group (clang-23 form): zero
  __builtin_amdgcn_tensor_load_to_lds(g0, g1, g2, g3, g4, 0);
}
#endif

// Nonzero span of padded anti-diagonal q: rows i in [i0, i1], first element
// of the span is D[b, i0-1, q-i0-1], global stride (Mm-1) floats per row.
__device__ __forceinline__ void diag_span(int q, int& i0, int& i1, int& L) {
  i0 = (q - Mm > 1) ? (q - Mm) : 1;
  i1 = (q - 1 < Nn) ? (q - 1) : Nn;
  L  = i1 - i0 + 1;
}

// One block per batch; 512 threads (16 wave32). Thread t owns row i = t+1.
// LDS rings:
//   dRing/sRing: 5-deep ring of Dp/Sp anti-diagonals. Iter d reads slots
//                {d,d-1,d-2}%5 while slot (d+1)%5 is in flight and slot
//                (d+2)%5 is the new TDM target -- all distinct mod 5.
//   rR/rS:       3-deep ring of R/SigR anti-diagonals (DP state)
__global__ __launch_bounds__(Nn) void udtw_fwd(const float* __restrict__ D,
                                               const float* __restrict__ S,
                                               float* __restrict__ out) {
  __shared__ float dRing[5][Nn + 1];
  __shared__ float sRing[5][Nn + 1];
  __shared__ float rR[3][Nn + 1];
  __shared__ float rS[3][Nn + 1];

  const int t = (int)threadIdx.x;
  const int b = (int)blockIdx.x;
  const int i = t + 1;                       // 1..Nn
  const size_t base = (size_t)b * Nn * Mm;

  // Init diag rows 0 (all BIG except [0]=0) and 1 (all BIG), as in reference.
  for (int k = t; k <= Nn; k += Nn) {        // t=0 covers k=0 and k=512
    const float v0 = (k == 0) ? 0.0f : BIGF;
    rR[0][k] = v0;   rS[0][k] = v0;
    rR[1][k] = BIGF; rS[1][k] = BIGF;
  }

  // Prologue: stage diagonals q=2 and q=3 into slots 2 and 3.
#if UDTW_HAVE_TDM
  if (t == 0) {
    for (int q = 2; q <= 3; ++q) {
      int i0, i1, L; diag_span(q, i0, i1, L);
      const size_t g0 = base + (size_t)(i0 - 1) * Mm + (size_t)(q - i0 - 1);
      tdm_diag_load(D + g0, lds_off(&dRing[q % 5][i0]), L);
      tdm_diag_load(S + g0, lds_off(&sRing[q % 5][i0]), L);
    }
    __builtin_amdgcn_s_wait_tensorcnt(2);   // diag-2 pair done; diag-3 in flight
  }
#else
  for (int q = 2; q <= 3; ++q) {
    int i0, i1, L; diag_span(q, i0, i1, L);
    if (t < L) {
      const size_t g = base + (size_t)(i0 - 1) * Mm + (size_t)(q - i0 - 1)
                     + (size_t)t * (Mm - 1);
      dRing[q % 5][i0 + t] = D[g];
      sRing[q % 5][i0 + t] = S[g];
    }
  }
#endif
  __syncthreads();

  for (int d = 2; d <= DMAX; ++d) {
    // ---- Prefetch diagonal d+2 into slot (d+2)%5 ------------------------
    // Its previous contents (diag d-3) were last read at step d-1, which
    // the trailing barrier of step d-1 has already ordered.
    const int q = d + 2;
    if (q <= DMAX) {
      int i0, i1, L; diag_span(q, i0, i1, L);
      const size_t g0 = base + (size_t)(i0 - 1) * Mm + (size_t)(q - i0 - 1);
      const int slot = q % 5;
#if UDTW_HAVE_TDM
      if (t == 0) {
        tdm_diag_load(D + g0, lds_off(&dRing[slot][i0]), L);
        tdm_diag_load(S + g0, lds_off(&sRing[slot][i0]), L);
      }
#else
      if (t < L) {
        const size_t g = g0 + (size_t)t * (Mm - 1);
        dRing[slot][i0 + t] = D[g];
        sRing[slot][i0 + t] = S[g];
      }
#endif
    }

    // ---- Compute diagonal d ---------------------------------------------
    const int j  = d - i;
    const int s0 = d % 3, s1 = (d + 2) % 3, s2 = (d + 1) % 3; // d, d-1, d-2 mod 3
    float nR = BIGF, nS = BIGF;
    if (j >= 1 && j <= Mm) {
      // softmin weights from R neighbors (alpha = 1/GAMMA = 1)
      const float rr0 = -rR[s2][i - 1];      // R(i-1, j-1)   diag d-2
      const float rr1 = -rR[s1][i - 1];      // R(i-1, j)     diag d-1
      const float rr2 = -rR[s1][i];          // R(i,   j-1)   diag d-1
      const float rmx = fmaxf(rr0, fmaxf(rr1, rr2));
      const float e0  = expf(rr0 - rmx);
      const float e1  = expf(rr1 - rmx);
      const float e2  = expf(rr2 - rmx);
      const float inv = 1.0f / (e0 + e1 + e2);
      const float p0 = e0 * inv, p1 = e1 * inv, p2 = e2 * inv;

      const int q0 = d % 5, q1 = (d + 4) % 5, q2 = (d + 3) % 5; // d, d-1, d-2 mod 5
      const bool gi = (i >= 2), gj = (j >= 2);   // guards keep reads in loaded span
      const float dA = dRing[q0][i];                               // Dp(i, j)
      const float dB = (gi && gj) ? dRing[q2][i - 1] : 0.0f;       // Dp(i-1, j-1)
      const float dC = gi ? dRing[q1][i - 1] : 0.0f;               // Dp(i-1, j)
      const float dE = gj ? dRing[q1][i] : 0.0f;                   // Dp(i,   j-1)
      nR = dA + p0 * dB + p1 * dC + p2 * dE;

      const float sA = sRing[q0][i];
      const float sB = (gi && gj) ? sRing[q2][i - 1] : 0.0f;
      const float sC = gi ? sRing[q1][i - 1] : 0.0f;
      const float sE = gj ? sRing[q1][i] : 0.0f;
      nS = sA + p0 * sB + p1 * sC + p2 * sE;
    }
    rR[s0][i] = nR;
    rS[s0][i] = nS;
    if (t == 0) { rR[s0][0] = BIGF; rS[s0][0] = BIGF; }  // row-0 boundary
    if (d == DMAX && i == Nn) { out[b] = nR; out[Bb + b] = nS; }

#if UDTW_HAVE_TDM
    // TDM ops from one wave complete in order: <=2 outstanding means the
    // diag d+1 pair (needed next iteration) has landed in LDS.
    if (t == 0) {
      if (q <= DMAX) __builtin_amdgcn_s_wait_tensorcnt(2);
      else           __builtin_amdgcn_s_wait_tensorcnt(0);
    }
#endif
    __syncthreads();   // publish diag d (R/S) and prefetched D/Sig diagonals
  }
}

extern "C" void kernel_launch(void* const* d_in, const int* in_sizes, int n_in,
                              void* d_out, int out_size, void* d_ws, size_t ws_size,
                              hipStream_t stream) {
  (void)in_sizes; (void)n_in; (void)out_size; (void)d_ws; (void)ws_size;
  const float* D  = (const float*)d_in[0];
  const float* Sg = (const float*)d_in[1];
  float* out = (float*)d_out;
  hipLaunchKernelGGL(udtw_fwd, dim3(Bb), dim3(Nn), 0, stream, D, Sg, out);
}